// VRWKV_SpatialMix_43731357008655
// MI455X (gfx1250) — compile-verified
//
#include <hip/hip_runtime.h>
#include <math.h>

// ---------------------------------------------------------------------------
// VRWKV SpatialMix for MI455X (gfx1250, wave32, WMMA).
// Memory-bound workload (>10x vs any matrix ceiling at 23.3 TB/s), so GEMMs
// use full-precision V_WMMA_F32_16X16X4_F32 (no bf16 precision loss needed).
// Both GEMM operands are staged into LDS (X cooperatively, W per-wave) via
// GLOBAL_LOAD_ASYNC_TO_LDS_B128 (ASYNCcnt) so the inner loop is pure
// ds_load + v_wmma instead of latency-bound strided global b64s.
// Workspace use: 3*BTC + B*T + 12*B*NC*C floats (~120 MB) — assumed <= ws_size.
// ---------------------------------------------------------------------------

typedef __attribute__((ext_vector_type(2))) float v2f;
typedef __attribute__((ext_vector_type(8))) float v8f;
typedef int v4i __attribute__((vector_size(16)));   // pointee type of the
                                                    // async-to-LDS builtin

#define AS1 __attribute__((address_space(1)))
#define AS3 __attribute__((address_space(3)))

#if __has_builtin(__builtin_amdgcn_global_load_async_to_lds_b128)
#define HAVE_ASYNC_LDS 1
#else
#define HAVE_ASYNC_LDS 0
#endif

constexpr int Bn = 8, Hh = 64, Ww = 64, C = 256;
constexpr int T  = Hh * Ww;          // 4096
constexpr int BT = Bn * T;           // 32768
constexpr int CL = 16;               // scan chunk length
constexpr int NC = T / CL;           // 256 chunks per (b,c) lane
constexpr float INV_T = 1.0f / (float)T;

__device__ __forceinline__ v8f wmma_f32(v2f a, v2f b, v8f c) {
  // 8-arg form: (neg_a, A, neg_b, B, c_mod, C, reuse_a, reuse_b)
  return __builtin_amdgcn_wmma_f32_16x16x4_f32(false, a, false, b, (short)0, c,
                                               false, false);
}

// 16B global -> LDS copy: async (ASYNCcnt) when available, else sync float4.
__device__ __forceinline__ void stage16(const float* __restrict__ g, float* l) {
#if HAVE_ASYNC_LDS
  __builtin_amdgcn_global_load_async_to_lds_b128((AS1 v4i*)g, (AS3 v4i*)l, 0,
                                                 0);
#else
  *(float4*)l = *(const float4*)g;
#endif
}
__device__ __forceinline__ void stage_wait() {
#if HAVE_ASYNC_LDS
#if __has_builtin(__builtin_amdgcn_s_wait_asynccnt)
  __builtin_amdgcn_s_wait_asynccnt(0);
#else
  asm volatile("s_wait_asynccnt 0x0" ::: "memory");
#endif
#endif
}

// ---------------------------------------------------------------------------
// Y = X @ W.T   (M=BT, N=C=256, K=C=256), fp32 WMMA 16x16x4.
// Block: 256 thr = 8 waves. M-tile 64 rows; waves = 4 M-subtiles x 2 N-halves.
// K-chunks of 64, LDS pitch 68 (pitch%64==4, even -> the 16x4 fragment v2f
// reads of lanes 0-15 / 16-31 land on 32 distinct banks, conflict-free).
// NQ==3: Y2 gets sigmoid applied (r -> sr).
// ---------------------------------------------------------------------------
template <int NQ>
__global__ __launch_bounds__(256) void gemm_xwt(
    const float* __restrict__ X,
    const float* __restrict__ W0, const float* __restrict__ W1,
    const float* __restrict__ W2,
    float* __restrict__ Y0, float* __restrict__ Y1, float* __restrict__ Y2) {
  constexpr int KC = 64, KCP = 68, NKC = C / KC;
  __shared__ float Xs[64 * KCP];          // 17,408 B
  __shared__ float Ws[8 * 16 * KCP];      // 34,816 B (per-wave 16-row regions)

  const int tid  = threadIdx.x;
  const int lane = tid & 31;
  const int wv   = tid >> 5;   // 0..7
  const int wm   = wv & 3;     // M sub-tile (16 rows each)
  const int wn   = wv >> 2;    // N half (128 cols each)
  const int m0   = blockIdx.x * 64;
  const int lr   = lane & 15;  // row/col within fragment
  const int lh   = lane >> 4;  // K-pair select (0 -> K{0,1}, 1 -> K{2,3})
  float* Wsw = Ws + wv * 16 * KCP;

  for (int q = 0; q < NQ; ++q) {
    const float* Wq = (q == 0) ? W0 : (q == 1) ? W1 : W2;
    float* Yq       = (q == 0) ? Y0 : (q == 1) ? Y1 : Y2;

    // Warm weight rows into L2 (global_prefetch_b8), 1 row per thread.
    __builtin_prefetch(Wq + (size_t)tid * C, 0, 1);

    v8f acc[8];
#pragma unroll
    for (int nt = 0; nt < 8; ++nt) { v8f z = {}; acc[nt] = z; }

    for (int kc = 0; kc < NKC; ++kc) {
      // Stage X tile: 64 rows x 64 K, coalesced b128, 4 copies/thread.
#pragma unroll
      for (int it = 0; it < 4; ++it) {
        int idx = it * 256 + tid;           // 1024 float4 slots
        int row = idx >> 4;                 // 16 float4 per row
        int c4  = idx & 15;
        stage16(X + (size_t)(m0 + row) * C + kc * KC + c4 * 4,
                &Xs[row * KCP + c4 * 4]);
      }
      stage_wait();
      __syncthreads();

#pragma unroll
      for (int nt = 0; nt < 8; ++nt) {
        const int n0 = wn * 128 + nt * 16;
        // Stage this wave's 16x64 W tile (wave-private region, no barrier):
        // 256 float4, 8 copies/lane, 2x256B segments per instruction.
#pragma unroll
        for (int it = 0; it < 8; ++it) {
          int idx = it * 32 + lane;
          int row = idx >> 4;
          int c4  = idx & 15;
          stage16(Wq + (size_t)(n0 + row) * C + kc * KC + c4 * 4,
                  &Wsw[row * KCP + c4 * 4]);
        }
        stage_wait();

        const float* Arow = Xs + (wm * 16 + lr) * KCP + lh * 2;
        const float* Brow = Wsw + lr * KCP + lh * 2;
#pragma unroll
        for (int kk = 0; kk < 16; ++kk) {   // K steps of 4
          v2f a = *(const v2f*)(Arow + kk * 4);
          v2f b = *(const v2f*)(Brow + kk * 4);
          acc[nt] = wmma_f32(a, b, acc[nt]);
        }
      }
      __syncthreads();
    }

    // Epilogue: C/D layout — VGPR i, lane l -> row i + 8*(l>>4), col l&15.
#pragma unroll
    for (int nt = 0; nt < 8; ++nt) {
      const int n0 = wn * 128 + nt * 16;
#pragma unroll
      for (int i = 0; i < 8; ++i) {
        float val = acc[nt][i];
        if (NQ == 3 && q == 2) val = 1.0f / (1.0f + expf(-val));  // sigmoid(r)
        int row = m0 + wm * 16 + i + lh * 8;
        Yq[(size_t)row * C + n0 + lr] = val;
      }
    }
  }
}

// ---------------------------------------------------------------------------
// Sobel magnitude, mean over channels, sigmoid -> edge[b, t].
// Block per (b, h); thread = channel; sliding 3x3 window, zero 'SAME' padding.
// ---------------------------------------------------------------------------
__global__ __launch_bounds__(256) void edge_gate(const float* __restrict__ X,
                                                 float* __restrict__ edge) {
  __shared__ float part[8][64];
  const int b = blockIdx.x / Hh;
  const int h = blockIdx.x % Hh;
  const int c = threadIdx.x;
  const int lane = c & 31, wvi = c >> 5;

  const float* xb = X + (size_t)b * T * C + c;
  const bool hm = (h > 0), hp = (h < Hh - 1);
  const float* rm = xb + (size_t)(h - 1) * Ww * C;
  const float* rc = xb + (size_t)h * Ww * C;
  const float* rp = xb + (size_t)(h + 1) * Ww * C;

  // columns w-1 (a), w (b), w+1 (c); rows h-1(0), h(1), h+1(2)
  float a0 = 0.f, a1 = 0.f, a2 = 0.f;
  float b0 = hm ? rm[0] : 0.f, b1 = rc[0], b2 = hp ? rp[0] : 0.f;
  float c0 = hm ? rm[C] : 0.f, c1 = rc[C], c2 = hp ? rp[C] : 0.f;

  for (int w = 0; w < Ww; ++w) {
    float gx = (c0 - a0) + 2.f * (c1 - a1) + (c2 - a2);
    float gy = (a2 + 2.f * b2 + c2) - (a0 + 2.f * b0 + c0);
    float mag = sqrtf(gx * gx + gy * gy + 1e-8f);
#pragma unroll
    for (int off = 16; off; off >>= 1) mag += __shfl_xor(mag, off, 32);
    if (lane == 0) part[wvi][w] = mag;
    a0 = b0; a1 = b1; a2 = b2;
    b0 = c0; b1 = c1; b2 = c2;
    int w2 = w + 2;
    if (w2 < Ww) {
      size_t off2 = (size_t)w2 * C;
      c0 = hm ? rm[off2] : 0.f; c1 = rc[off2]; c2 = hp ? rp[off2] : 0.f;
    } else {
      c0 = c1 = c2 = 0.f;
    }
  }
  __syncthreads();
  if (c < Ww) {
    float s = 0.f;
#pragma unroll
    for (int i = 0; i < 8; ++i) s += part[i][c];
    s *= (1.0f / (float)C);
    edge[(size_t)b * T + h * Ww + c] = 1.0f / (1.0f + expf(-s));
  }
}

// ---------------------------------------------------------------------------
// Stabilized WKV recurrence helpers. State (a,b,p) represents (a*e^p, b*e^p).
// ---------------------------------------------------------------------------
__device__ __forceinline__ void wkv_step(float& a, float& b, float& p, float kk,
                                         float vv, float w) {
  float ww = p - w;
  float pp = fmaxf(ww, kk);
  float e1 = expf(ww - pp), e2 = expf(kk - pp);
  a = e1 * a + e2 * vv;
  b = e1 * b + e2;
  p = pp;
}
// carry := decay(carry, L*w) combined with chunk summary q (zero-init fold).
__device__ __forceinline__ void wkv_combine(float& a, float& b, float& p,
                                            float qa, float qb, float qp,
                                            float dL) {
  float p1 = p - dL;
  float m  = fmaxf(p1, qp);
  float e1 = expf(p1 - m), e2 = expf(qp - m);
  a = e1 * a + e2 * qa;
  b = e1 * b + e2 * qb;
  p = m;
}

// Phase 1: per (b, chunk, c) fwd+bwd chunk summaries from zero state.
__global__ __launch_bounds__(256) void wkv_phase1(
    const float* __restrict__ k, const float* __restrict__ v,
    const float* __restrict__ decay,
    float* __restrict__ sFa, float* __restrict__ sFb, float* __restrict__ sFp,
    float* __restrict__ sBa, float* __restrict__ sBb, float* __restrict__ sBp) {
  const int g = blockIdx.x * 256 + threadIdx.x;  // (b*NC + j)*C + c
  const int c = g % C;
  const int j = (g / C) % NC;
  const int b = g / (C * NC);
  const float w = decay[c] * INV_T;
  const size_t base = ((size_t)b * T + (size_t)j * CL) * C + c;

  float kr[CL], vr[CL];
#pragma unroll
  for (int i = 0; i < CL; ++i) {
    kr[i] = k[base + (size_t)i * C];
    vr[i] = v[base + (size_t)i * C];
  }
  float a = 0.f, bb = 0.f, p = -1e38f;
#pragma unroll
  for (int i = 0; i < CL; ++i) wkv_step(a, bb, p, kr[i], vr[i], w);
  sFa[g] = a; sFb[g] = bb; sFp[g] = p;
  a = 0.f; bb = 0.f; p = -1e38f;
#pragma unroll
  for (int i = CL - 1; i >= 0; --i) wkv_step(a, bb, p, kr[i], vr[i], w);
  sBa[g] = a; sBb[g] = bb; sBp[g] = p;
}

// Phase 2: per (b, c) exclusive scan over chunk summaries -> carry-in states.
__global__ __launch_bounds__(256) void wkv_phase2(
    const float* __restrict__ decay,
    const float* __restrict__ sFa, const float* __restrict__ sFb,
    const float* __restrict__ sFp, const float* __restrict__ sBa,
    const float* __restrict__ sBb, const float* __restrict__ sBp,
    float* __restrict__ cFa, float* __restrict__ cFb, float* __restrict__ cFp,
    float* __restrict__ cBa, float* __restrict__ cBb, float* __restrict__ cBp) {
  const int g = blockIdx.x * 256 + threadIdx.x;  // b*C + c
  const int c = g % C;
  const int b = g / C;
  const float w  = decay[c] * INV_T;
  const float dL = (float)CL * w;

  float a = 0.f, bb = 0.f, p = -1e38f;
  for (int j = 0; j < NC; ++j) {
    size_t idx = ((size_t)b * NC + j) * C + c;
    cFa[idx] = a; cFb[idx] = bb; cFp[idx] = p;
    wkv_combine(a, bb, p, sFa[idx], sFb[idx], sFp[idx], dL);
  }
  a = 0.f; bb = 0.f; p = -1e38f;
  for (int j = NC - 1; j >= 0; --j) {
    size_t idx = ((size_t)b * NC + j) * C + c;
    cBa[idx] = a; cBb[idx] = bb; cBp[idx] = p;
    wkv_combine(a, bb, p, sBa[idx], sBb[idx], sBp[idx], dL);
  }
}

// Phase 3: per chunk, replay bwd states into registers, sweep fwd, combine
// with ps = u + k, and fuse gating: sr <- (sr + edge) * xwkv  (in place).
__global__ __launch_bounds__(256) void wkv_phase3(
    const float* __restrict__ k, const float* __restrict__ v,
    const float* __restrict__ decay, const float* __restrict__ first,
    const float* __restrict__ cFa, const float* __restrict__ cFb,
    const float* __restrict__ cFp, const float* __restrict__ cBa,
    const float* __restrict__ cBb, const float* __restrict__ cBp,
    const float* __restrict__ edge, float* __restrict__ sr) {
  const int g = blockIdx.x * 256 + threadIdx.x;  // (b*NC + j)*C + c
  const int c = g % C;
  const int j = (g / C) % NC;
  const int b = g / (C * NC);
  const int t0 = j * CL;
  const float w = decay[c] * INV_T;
  const float u = first[c] * INV_T;
  const size_t base = ((size_t)b * T + t0) * C + c;

  float kr[CL], vr[CL];
#pragma unroll
  for (int i = 0; i < CL; ++i) {
    kr[i] = k[base + (size_t)i * C];
    vr[i] = v[base + (size_t)i * C];
  }
  // Backward states (exclusive of element i: elements t > i).
  float ra[CL], rb[CL], rp[CL];
  float a = cBa[g], bb = cBb[g], p = cBp[g];
#pragma unroll
  for (int i = CL - 1; i >= 0; --i) {
    ra[i] = a; rb[i] = bb; rp[i] = p;
    wkv_step(a, bb, p, kr[i], vr[i], w);
  }
  // Forward sweep (exclusive) + combine + fused gating.
  a = cFa[g]; bb = cFb[g]; p = cFp[g];
#pragma unroll
  for (int i = 0; i < CL; ++i) {
    float ps = u + kr[i];
    float m  = fmaxf(fmaxf(p, rp[i]), ps);
    float ef = expf(p - m), eb = expf(rp[i] - m), es = expf(ps - m);
    float num = a * ef + ra[i] * eb + es * vr[i];
    float den = bb * ef + rb[i] * eb + es;
    size_t oi = base + (size_t)i * C;
    float gate = sr[oi] + edge[(size_t)b * T + t0 + i];
    sr[oi] = gate * (num / den);
    wkv_step(a, bb, p, kr[i], vr[i], w);
  }
}

// ---------------------------------------------------------------------------
extern "C" void kernel_launch(void* const* d_in, const int* in_sizes, int n_in,
                              void* d_out, int out_size, void* d_ws,
                              size_t ws_size, hipStream_t stream) {
  (void)in_sizes; (void)n_in; (void)out_size; (void)ws_size;
  const float* x   = (const float*)d_in[0];
  const float* Wk  = (const float*)d_in[1];
  const float* Wv  = (const float*)d_in[2];
  const float* Wr  = (const float*)d_in[3];
  const float* Wo  = (const float*)d_in[4];
  const float* dec = (const float*)d_in[5];
  const float* fst = (const float*)d_in[6];

  float* f = (float*)d_ws;
  const size_t BTC = (size_t)BT * C;
  float* kbuf = f;               // k = x @ Wk.T
  float* vbuf = kbuf + BTC;      // v = x @ Wv.T
  float* sbuf = vbuf + BTC;      // sigmoid(r), later overwritten with gated
  float* ebuf = sbuf + BTC;      // edge gate (B*T)
  const size_t S = (size_t)Bn * NC * C;
  float* sFa = ebuf + (size_t)Bn * T;
  float* sFb = sFa + S; float* sFp = sFb + S;
  float* sBa = sFp + S; float* sBb = sBa + S; float* sBp = sBb + S;
  float* cFa = sBp + S; float* cFb = cFa + S; float* cFp = cFb + S;
  float* cBa = cFp + S; float* cBb = cBa + S; float* cBp = cBb + S;

  dim3 blk(256);
  gemm_xwt<3><<<BT / 64, blk, 0, stream>>>(x, Wk, Wv, Wr, kbuf, vbuf, sbuf);
  edge_gate<<<Bn * Hh, blk, 0, stream>>>(x, ebuf);
  wkv_phase1<<<(Bn * NC * C) / 256, blk, 0, stream>>>(kbuf, vbuf, dec, sFa, sFb,
                                                      sFp, sBa, sBb, sBp);
  wkv_phase2<<<(Bn * C) / 256, blk, 0, stream>>>(dec, sFa, sFb, sFp, sBa, sBb,
                                                 sBp, cFa, cFb, cFp, cBa, cBb,
                                                 cBp);
  wkv_phase3<<<(Bn * NC * C) / 256, blk, 0, stream>>>(
      kbuf, vbuf, dec, fst, cFa, cFb, cFp, cBa, cBb, cBp, ebuf, sbuf);
  gemm_xwt<1><<<BT / 64, blk, 0, stream>>>(sbuf, Wo, nullptr, nullptr,
                                           (float*)d_out, nullptr, nullptr);
}